// FastNeuron_55078660604153
// MI455X (gfx1250) — compile-verified
//
#include <hip/hip_runtime.h>
#include <hip/hip_bf16.h>
#include <math.h>

// ---------------------------------------------------------------------------
// FastNeuron step for MI455X (gfx1250), wave32.
// Dense layers: V_WMMA_F32_16X16X4_F32 with a branchless, software-pipelined
// K-loop (A pre-packed/zero-padded, B column-clamped, EXEC all-1s throughout).
// Bandwidth-bound parts (2 passes over h, ~1 GB @ 23.3 TB/s => ~43us floor)
// are coalesced streaming kernels with block LN reductions.
// ---------------------------------------------------------------------------

typedef __attribute__((ext_vector_type(2))) float v2f;
typedef __attribute__((ext_vector_type(8))) float v8f;

static constexpr int B  = 8;
static constexpr int S  = 4096;
static constexpr int D  = 2560;
static constexpr int R  = 64;
static constexpr int DQ = 128;
static constexpr int DV = 512;
static constexpr int DP = 128;
static constexpr int DC = 128;
static constexpr int DH = 384;

static constexpr size_t HNEW_N = (size_t)B * S * D;   // 83886080
static constexpr size_t MEM_N  = (size_t)B * D * R;   // 1310720
static constexpr size_t WDN_N  = (size_t)B * D * DP;  // 2621440
static constexpr size_t WUN_N  = (size_t)B * DP * D;  // 2621440

// -------------------- device helpers --------------------

__device__ __forceinline__ float gelu_f(float x) {
    return 0.5f * x * (1.0f + erff(x * 0.70710678118654752440f));
}

__device__ __forceinline__ float act_f(float x, int act) {
    switch (act) {
        case 1: return gelu_f(x);
        case 2: return 1.0f / (1.0f + expf(-x));                     // sigmoid
        case 3: { float sp = (x > 20.0f) ? x : log1pf(expf(x));      // min(softplus,.1)
                  return fminf(sp, 0.1f); }
        default: return x;
    }
}

__device__ __forceinline__ float block_reduce256(float v, float* red) {
    int t = threadIdx.x;
    red[t] = v;
    __syncthreads();
    for (int off = 128; off > 0; off >>= 1) {
        if (t < off) red[t] += red[t + off];
        __syncthreads();
    }
    float r = red[0];
    __syncthreads();
    return r;
}

// Up to 4 concatenated row-segments: X[row, k] gathered on the fly
// (used only by the cold pack_x kernel, never in the WMMA hot loop).
struct Seg4 {
    const float* p0; const float* p1; const float* p2; const float* p3;
    int l0, l1, l2, l3;   // segment lengths along K
    int s0, s1, s2, s3;   // per-row strides
};

__device__ __forceinline__ float seg_load(const Seg4& g, int row, int k) {
    if (k < g.l0) return g.p0[(size_t)row * g.s0 + k];
    k -= g.l0;
    if (k < g.l1) return g.p1[(size_t)row * g.s1 + k];
    k -= g.l1;
    if (k < g.l2) return g.p2[(size_t)row * g.s2 + k];
    k -= g.l2;
    return g.p3[(size_t)row * g.s3 + k];
}

// Pack concatenated activations into dense 16 x Kpad, zero-padded (cold path).
__global__ __launch_bounds__(256) void pack_x(
    Seg4 X, float* __restrict__ xp, int M, int K, int Kpad)
{
    int idx = blockIdx.x * 256 + threadIdx.x;
    if (idx >= 16 * Kpad) return;
    int row = idx / Kpad, k = idx % Kpad;
    xp[idx] = (row < M && k < K) ? seg_load(X, row, k) : 0.0f;
}

__global__ __launch_bounds__(256) void zero_f(float* __restrict__ p, int n)
{
    int i = blockIdx.x * 256 + threadIdx.x;
    if (i < n) p[i] = 0.0f;
}

// -------------------- WMMA GEMM: out[M,N] = act(X[M,K] @ W[K,N] + bias) -----
// One wave per 16x16 C tile; V_WMMA_F32_16X16X4_F32, pipelined K-loop.
// Contract on X: rows [0, gridDim.y*16) readable; cols [0, ceil4(K)) readable
// with cols >= K zero (or K % 4 == 0). Garbage rows >= Mstore are benign
// (row-local, never stored). EXEC stays all-1s through every WMMA.
// A frag (16x4 f32): lanes 0-15 hold K=kb,kb+1 ; lanes 16-31 hold kb+2,kb+3.
// B frag (4x16 f32): same K split; N = lane&15 for both halves.

__global__ __launch_bounds__(32) void gemm_wmma(
    const float* __restrict__ X, int ldx,
    const float* __restrict__ W, const float* __restrict__ bias,
    float* __restrict__ out, int Mstore, int K, int N, int act)
{
    const int lane = threadIdx.x;
    const int half = lane >> 4;     // 0 or 1
    const int l16  = lane & 15;
    const int n0   = blockIdx.x * 16;
    const int m0   = blockIdx.y * 16;
    const int n    = n0 + l16;
    const int ncl  = (n < N) ? n : (N - 1);   // clamp: loads always in-bounds
    const int K4   = K & ~3;

    const float* xq = X + (size_t)(m0 + l16) * ldx + 2 * half;
    const float* wq = W + (size_t)(2 * half) * N + ncl;

    v8f acc = {};
    const int nIter = K4 >> 2;
    if (nIter > 0) {
        v2f a0 = *(const v2f*)xq;
        v2f b0; b0.x = wq[0]; b0.y = wq[N];
        for (int it = 1; it < nIter; ++it) {
            xq += 4;
            wq += (size_t)4 * N;
            v2f a1 = *(const v2f*)xq;
            v2f b1; b1.x = wq[0]; b1.y = wq[N];
            __builtin_prefetch(wq + (size_t)16 * N, 0, 0);   // global_prefetch_b8
            acc = __builtin_amdgcn_wmma_f32_16x16x4_f32(
                false, a0, false, b0, (short)0, acc, false, false);
            a0 = a1; b0 = b1;
        }
        acc = __builtin_amdgcn_wmma_f32_16x16x4_f32(
            false, a0, false, b0, (short)0, acc, false, false);
    }
    if (K4 < K) {   // tail: A cols >= K are zero-padded; guard only B
        const int ka = K4 + 2 * half;
        v2f a = *(const v2f*)(X + (size_t)(m0 + l16) * ldx + K4 + 2 * half);
        v2f b;
        b.x = (ka     < K) ? W[(size_t)ka * N + ncl]       : 0.0f;
        b.y = (ka + 1 < K) ? W[(size_t)(ka + 1) * N + ncl] : 0.0f;
        acc = __builtin_amdgcn_wmma_f32_16x16x4_f32(
            false, a, false, b, (short)0, acc, false, false);
    }

    if (n < N) {
        const float bb = bias ? bias[n] : 0.0f;
#pragma unroll
        for (int v = 0; v < 8; ++v) {
            const int m = m0 + v + 8 * half;   // C/D layout: VGPR v -> M=v / v+8
            if (m < Mstore) out[(size_t)m * N + n] = act_f(acc[v] + bb, act);
        }
    }
}

// -------------------- h mean over S (2-stage, deterministic) ---------------

static constexpr int CHUNKS = 32;               // 4096 / 32 = 128 rows per chunk

__global__ __launch_bounds__(256) void havg_partial(
    const float* __restrict__ h, float* __restrict__ part)
{
    const int col   = blockIdx.x * 256 + threadIdx.x;   // 0 .. B*D-1
    const int chunk = blockIdx.y;                        // 0 .. CHUNKS-1
    const int b = col / D, d = col % D;
    float s = 0.0f;
    size_t base = ((size_t)b * S + (size_t)chunk * (S / CHUNKS)) * D + d;
    for (int i = 0; i < S / CHUNKS; ++i) s += h[base + (size_t)i * D];
    part[(size_t)chunk * (B * D) + col] = s;
}

__global__ __launch_bounds__(256) void havg_final(
    const float* __restrict__ part, float* __restrict__ havg)
{
    const int col = blockIdx.x * 256 + threadIdx.x;
    float s = 0.0f;
    for (int c = 0; c < CHUNKS; ++c) s += part[(size_t)c * (B * D) + col];
    havg[col] = s * (1.0f / (float)S);
}

// -------------------- small elementwise pieces -----------------------------

__global__ __launch_bounds__(256) void vec_sub(
    const float* __restrict__ a, const float* __restrict__ b,
    float* __restrict__ c, int n)
{
    int i = blockIdx.x * 256 + threadIdx.x;
    if (i < n) c[i] = a[i] - b[i];
}

// mem = mem_A + lr[b] * wval[b,d] * wkey[b,r]; also per-block sumsq partial
__global__ __launch_bounds__(256) void mem_update(
    const float* __restrict__ memA, const float* __restrict__ lr,
    const float* __restrict__ wval, const float* __restrict__ wkey,
    float* __restrict__ memout, float* __restrict__ part)
{
    __shared__ float red[256];
    const int b = blockIdx.y;
    const int i = blockIdx.x * 256 + threadIdx.x;   // within-batch index, D*R
    const int d = i >> 6, r = i & 63;
    const size_t idx = (size_t)b * (D * R) + i;
    float v = memA[idx] + lr[b] * wval[(size_t)b * D + d] * wkey[b * R + r];
    memout[idx] = v;
    float ss = block_reduce256(v * v, red);
    if (threadIdx.x == 0) part[b * gridDim.x + blockIdx.x] = ss;
}

__global__ __launch_bounds__(256) void mem_norm(
    const float* __restrict__ part, int nblk, float* __restrict__ scl)
{
    __shared__ float red[256];
    const int b = blockIdx.x;
    float s = 0.0f;
    for (int i = threadIdx.x; i < nblk; i += 256) s += part[b * nblk + i];
    float tot = block_reduce256(s, red);
    if (threadIdx.x == 0) {
        float norm = sqrtf(tot);
        scl[b] = (norm > 10.0f) ? 10.0f / (norm + 1e-8f) : 1.0f;
    }
}

__global__ __launch_bounds__(256) void mem_scale(
    const float* __restrict__ scl, float* __restrict__ memout)
{
    const int b = blockIdx.y;
    const size_t idx = (size_t)b * (D * R) + blockIdx.x * 256 + threadIdx.x;
    memout[idx] *= scl[b];
}

// slots[b,r,:] = LN(mem[b,:,r]) * sln_w + sln_b   (one block per (b,r))
__global__ __launch_bounds__(256) void slots_ln(
    const float* __restrict__ mem, const float* __restrict__ w,
    const float* __restrict__ bb, float* __restrict__ slots)
{
    __shared__ float red[256];
    const int br = blockIdx.x;            // b*R + r
    const int b = br >> 6, r = br & 63;
    const int t = threadIdx.x;
    const float* mb = mem + (size_t)b * D * R;
    float x[D / 256];
    float s = 0.0f;
#pragma unroll
    for (int i = 0; i < D / 256; ++i) {
        int d = t + i * 256;
        x[i] = mb[(size_t)d * R + r];
        s += x[i];
    }
    float mean = block_reduce256(s, red) * (1.0f / (float)D);
    float vs = 0.0f;
#pragma unroll
    for (int i = 0; i < D / 256; ++i) { float c = x[i] - mean; vs += c * c; }
    float var = block_reduce256(vs, red) * (1.0f / (float)D);
    float rstd = rsqrtf(var + 1e-5f);
#pragma unroll
    for (int i = 0; i < D / 256; ++i) {
        int d = t + i * 256;
        slots[(size_t)br * D + d] = (x[i] - mean) * rstd * w[d] + bb[d];
    }
}

// scores -> softmax over R=64 -> mem_read_v[b, 0..511]
__global__ __launch_bounds__(128) void attention_k(
    const float* __restrict__ qry, const float* __restrict__ keys,
    const float* __restrict__ vals, float* __restrict__ mrv)
{
    __shared__ float attn[R];
    __shared__ float inv_s;
    const int b = blockIdx.x;
    const int t = threadIdx.x;
    if (t < R) {
        float s = 0.0f;
        const float* q = qry + b * DQ;
        const float* k = keys + ((size_t)b * R + t) * DQ;
        for (int i = 0; i < DQ; ++i) s += q[i] * k[i];
        attn[t] = s * 0.08838834764831843f;   // 1/sqrt(DQ)
    }
    __syncthreads();
    if (t == 0) {
        float mx = attn[0];
        for (int i = 1; i < R; ++i) mx = fmaxf(mx, attn[i]);
        float sum = 0.0f;
        for (int i = 0; i < R; ++i) { attn[i] = expf(attn[i] - mx); sum += attn[i]; }
        inv_s = 1.0f / sum;
    }
    __syncthreads();
    const float inv = inv_s;
    for (int v = t; v < DV; v += 128) {
        float s = 0.0f;
        for (int r = 0; r < R; ++r) s += attn[r] * vals[((size_t)b * R + r) * DV + v];
        mrv[b * DV + v] = s * inv;
    }
}

// down[b,p] = gelu( sum_d mem_read[b,d] * (W_down_base[d,p] + W_down_mod[b,d,p]) )
__global__ __launch_bounds__(128) void down_proj(
    const float* __restrict__ mrd, const float* __restrict__ wdb,
    const float* __restrict__ wdm, float* __restrict__ down)
{
    const int b = blockIdx.x, p = threadIdx.x;
    const float* m  = mrd + b * D;
    const float* wm = wdm + (size_t)b * D * DP;
    float s = 0.0f;
    for (int d = 0; d < D; ++d)
        s += m[d] * (wdb[(size_t)d * DP + p] + wm[(size_t)d * DP + p]);
    down[b * DP + p] = gelu_f(s);
}

// output[b,d] = 0.5*mem_read + sum_p down[b,p]*(W_up_base[p,d]+W_up_mod[b,p,d])
__global__ __launch_bounds__(256) void up_out(
    const float* __restrict__ down, const float* __restrict__ wub,
    const float* __restrict__ wum, const float* __restrict__ mrd,
    float* __restrict__ outv)
{
    const int b = blockIdx.y;
    const int d = blockIdx.x * 256 + threadIdx.x;
    const float* dn = down + b * DP;
    const float* wm = wum + (size_t)b * DP * D;
    float s = 0.0f;
    for (int p = 0; p < DP; ++p)
        s += dn[p] * (wub[(size_t)p * D + d] + wm[(size_t)p * D + d]);
    outv[b * D + d] = 0.5f * mrd[b * D + d] + s;
}

__global__ void calc_sc(const float* __restrict__ surp,
                        const float* __restrict__ plr, float* __restrict__ sc)
{
    int b = threadIdx.x;
    if (b < B) {
        float ws = 1.0f / (1.0f + expf(-10.0f * (surp[b] - 0.5f)));
        sc[b] = ws * plr[b];
    }
}

// W_down_new[b,d,p] = W_down_mod + sc[b]*d_pat[b,d]*d_addr[b,p]
__global__ __launch_bounds__(256) void wdn_k(
    const float* __restrict__ wdm, const float* __restrict__ sc,
    const float* __restrict__ rawd, float* __restrict__ out)
{
    const size_t idx = (size_t)blockIdx.x * 256 + threadIdx.x;   // B*D*DP
    const int b = (int)(idx / ((size_t)D * DP));
    const int rem = (int)(idx % ((size_t)D * DP));
    const int d = rem >> 7, p = rem & (DP - 1);
    out[idx] = wdm[idx] + sc[b] * rawd[b * (D + DP) + d] * rawd[b * (D + DP) + D + p];
}

// W_up_new[b,p,d] = W_up_mod + sc[b]*u_pat[b,p]*u_addr[b,d]
__global__ __launch_bounds__(256) void wun_k(
    const float* __restrict__ wum, const float* __restrict__ sc,
    const float* __restrict__ rawu, float* __restrict__ out)
{
    const size_t idx = (size_t)blockIdx.x * 256 + threadIdx.x;   // B*DP*D
    const int b = (int)(idx / ((size_t)DP * D));
    const int rem = (int)(idx % ((size_t)DP * D));
    const int p = rem / D, d = rem % D;
    out[idx] = wum[idx] + sc[b] * rawu[b * (DP + D) + p] * rawu[b * (DP + D) + DP + d];
}

// h_new[b,s,:] = LN(h[b,s,:] + g[b]*output[b,:]) * ln_w + ln_b  (one block/row)
__global__ __launch_bounds__(256) void final_ln(
    const float* __restrict__ h, const float* __restrict__ outv,
    const float* __restrict__ g, const float* __restrict__ lnw,
    const float* __restrict__ lnb, float* __restrict__ hnew)
{
    __shared__ float red[256];
    const int row = blockIdx.x;        // 0 .. B*S-1
    const int b = row >> 12;           // row / S
    const int t = threadIdx.x;
    const float gb = g[b];
    const size_t base = (size_t)row * D;
    float x[D / 256];
    float s = 0.0f;
#pragma unroll
    for (int i = 0; i < D / 256; ++i) {
        int d = t + i * 256;
        x[i] = h[base + d] + gb * outv[b * D + d];
        s += x[i];
    }
    float mean = block_reduce256(s, red) * (1.0f / (float)D);
    float vs = 0.0f;
#pragma unroll
    for (int i = 0; i < D / 256; ++i) { float c = x[i] - mean; vs += c * c; }
    float var = block_reduce256(vs, red) * (1.0f / (float)D);
    float rstd = rsqrtf(var + 1e-5f);
#pragma unroll
    for (int i = 0; i < D / 256; ++i) {
        int d = t + i * 256;
        hnew[base + d] = (x[i] - mean) * rstd * lnw[d] + lnb[d];
    }
}

// -------------------- host side --------------------------------------------

static inline Seg4 seg1(const float* p, int len, int str) {
    return Seg4{p, p, p, p, len, 0, 0, 0, str, 0, 0, 0};
}
static inline Seg4 seg2(const float* p0, int l0, int s0,
                        const float* p1, int l1, int s1) {
    return Seg4{p0, p1, p1, p1, l0, l1, 0, 0, s0, s1, 0, 0};
}
static inline Seg4 seg3(const float* p0, int l0, int s0,
                        const float* p1, int l1, int s1,
                        const float* p2, int l2, int s2) {
    return Seg4{p0, p1, p2, p2, l0, l1, l2, 0, s0, s1, s2, 0};
}
static inline Seg4 seg4v(const float* p0, int l0, int s0,
                         const float* p1, int l1, int s1,
                         const float* p2, int l2, int s2,
                         const float* p3, int l3, int s3) {
    return Seg4{p0, p1, p2, p3, l0, l1, l2, l3, s0, s1, s2, s3};
}

static inline int ceil4(int k) { return (k + 3) & ~3; }

extern "C" void kernel_launch(void* const* d_in, const int* in_sizes, int n_in,
                              void* d_out, int out_size, void* d_ws, size_t ws_size,
                              hipStream_t stream)
{
    // Inputs in setup_inputs() dict order; params dict flattened in insertion
    // order after the 6 tensor inputs.
    auto P = [&](int i) { return (const float*)d_in[i]; };
    const float* h    = P(0);
    const float* memA = P(1);
    const float* wdm  = P(2);
    const float* wum  = P(3);
    const float* pha  = P(4);
    const float* ctx  = P(5);
    const float *sp1_w = P(6),  *sp1_b = P(7),  *sp2_w = P(8),  *sp2_b = P(9);
    const float *su1_w = P(10), *su1_b = P(11), *su2_w = P(12), *su2_b = P(13);
    const float *wk1_w = P(14), *wk1_b = P(15), *wk2_w = P(16), *wk2_b = P(17);
    const float *wv1_w = P(18), *wv1_b = P(19), *wv2_w = P(20), *wv2_b = P(21);
    const float *lr1_w = P(22), *lr1_b = P(23), *lr2_w = P(24), *lr2_b = P(25);
    const float *sln_w = P(26), *sln_b = P(27);
    const float *rq_w  = P(28), *rq_b  = P(29);
    const float *W_K   = P(30), *W_V   = P(31);
    const float *vup_w = P(32), *vup_b = P(33);
    const float *wdb   = P(34), *wub   = P(35);
    const float *pd1_w = P(36), *pd1_b = P(37), *pd2_w = P(38), *pd2_b = P(39);
    const float *pu1_w = P(40), *pu1_b = P(41), *pu2_w = P(42), *pu2_b = P(43);
    const float *pl1_w = P(44), *pl1_b = P(45), *pl2_w = P(46), *pl2_b = P(47);
    const float *g1_w  = P(48), *g1_b  = P(49), *g2_w  = P(50), *g2_b  = P(51);
    const float *ln_w  = P(52), *ln_b  = P(53);
    // r1/r2 (54..57) unused by the reference forward.

    float* out    = (float*)d_out;
    float* hnew   = out;
    float* memout = out + HNEW_N;
    float* wdnout = memout + MEM_N;
    float* wunout = wdnout + WDN_N;

    // Big scratch: dead before final_ln runs, lives in the h_new region of
    // d_out (final_ln overwrites every element afterwards).
    float* PARTS = hnew;                                // CHUNKS*B*D = 655360
    float* SLOTS = PARTS + (size_t)CHUNKS * B * D;      // B*R*D     = 1310720
    float* KEYS  = SLOTS + (size_t)B * R * D;           // B*R*DQ    = 65536
    float* VALS  = KEYS  + (size_t)B * R * DQ;          // B*R*DV    = 262144

    // Small scratch (read by final kernels) lives in d_ws (~1.1 MB).
    float* wsbase = (float*)d_ws;
    float* w = wsbase;
    auto take = [&](size_t n) { float* r = w; w += (n + 63) & ~(size_t)63; return r; };
    const int PWIN = D + 1 + D + DC;                  // 5249, largest concat K
    float* XPAD = take((size_t)16 * ceil4(PWIN));     // packed A for concat GEMMs
    float* TBUF = take((size_t)16 * DH);              // 16 rows: rows 8..15 zero
    float* MRV  = take((size_t)16 * DV);              // 16 rows: rows 8..15 zero
    float* HAVG = take(B * D);
    float* PRED = take(B * D);
    float* ERR  = take(B * D);
    float* SURP = take(B);
    float* WKEY = take(B * R);
    float* WVAL = take(B * D);
    float* LRb  = take(B);
    float* SCL  = take(B);
    float* QRY  = take(B * DQ);
    float* MRD  = take(B * D);
    float* DOWN = take(B * DP);
    float* OUTV = take(B * D);
    float* RAWD = take(B * (D + DP));
    float* RAWU = take(B * (DP + D));
    float* PLR  = take(B);
    float* SCb  = take(B);
    float* Gb   = take(B);
    const int WS_N = (int)(w - wsbase);
    (void)n_in; (void)in_sizes; (void)out_size; (void)ws_size;

    // Deterministic scratch (zeros padding rows of TBUF/MRV etc.).
    zero_f<<<(WS_N + 255) / 256, 256, 0, stream>>>(wsbase, WS_N);

    auto pack = [&](Seg4 X, int K) {
        int Kp = ceil4(K);
        pack_x<<<(16 * Kp + 255) / 256, 256, 0, stream>>>(X, XPAD, B, K, Kp);
    };
    auto gemm = [&](const float* X, int ldx, const float* W, const float* bias,
                    float* O, int Mstore, int K, int N, int act) {
        dim3 g((N + 15) / 16, (Mstore + 15) / 16);
        gemm_wmma<<<g, 32, 0, stream>>>(X, ldx, W, bias, O, Mstore, K, N, act);
    };

    // 1) h_avg
    havg_partial<<<dim3((B * D) / 256, CHUNKS), 256, 0, stream>>>(h, PARTS);
    havg_final<<<(B * D) / 256, 256, 0, stream>>>(PARTS, HAVG);

    // 2) predicted = mlp([prev_h_avg, context], sp1, sp2);  error
    pack(seg2(pha, D, D, ctx, DC, DC), D + DC);
    gemm(XPAD, ceil4(D + DC), sp1_w, sp1_b, TBUF, B, D + DC, DH, 1);
    gemm(TBUF, DH, sp2_w, sp2_b, PRED, B, DH, D, 0);
    vec_sub<<<(B * D) / 256, 256, 0, stream>>>(HAVG, PRED, ERR, B * D);

    // 3) surprise = sigmoid(mlp([error, context], su1, su2))
    pack(seg2(ERR, D, D, ctx, DC, DC), D + DC);
    gemm(XPAD, ceil4(D + DC), su1_w, su1_b, TBUF, B, D + DC, DH, 1);
    gemm(TBUF, DH, su2_w, su2_b, SURP, B, DH, 1, 2);

    // 4) wkey / wval from wi = [h_avg, surprise, context] (pack once)
    const int WIN = D + 1 + DC;   // 2689
    pack(seg3(HAVG, D, D, SURP, 1, 1, ctx, DC, DC), WIN);
    gemm(XPAD, ceil4(WIN), wk1_w, wk1_b, TBUF, B, WIN, DH, 1);
    gemm(TBUF, DH, wk2_w, wk2_b, WKEY, B, DH, R, 0);
    gemm(XPAD, ceil4(WIN), wv1_w, wv1_b, TBUF, B, WIN, DH, 1);
    gemm(TBUF, DH, wv2_w, wv2_b, WVAL, B, DH, D, 0);

    // 5) lr = min(softplus(mlp([surprise, context], lr1, lr2)), 0.1)
    pack(seg2(SURP, 1, 1, ctx, DC, DC), 1 + DC);
    gemm(XPAD, ceil4(1 + DC), lr1_w, lr1_b, TBUF, B, 1 + DC, DH / 2, 1);
    gemm(TBUF, DH / 2, lr2_w, lr2_b, LRb, B, DH / 2, 1, 3);

    // 6) mem update + norm clamp (writes mem output directly)
    mem_update<<<dim3((D * R) / 256, B), 256, 0, stream>>>(memA, LRb, WVAL, WKEY, memout, PARTS);
    mem_norm<<<B, 256, 0, stream>>>(PARTS, (D * R) / 256, SCL);
    mem_scale<<<dim3((D * R) / 256, B), 256, 0, stream>>>(SCL, memout);

    // 7) slots = LN(mem^T); keys/vals GEMMs with M = B*R = 512 (K mult of 4)
    slots_ln<<<B * R, 256, 0, stream>>>(memout, sln_w, sln_b, SLOTS);
    gemm(SLOTS, D, W_K, nullptr, KEYS, B * R, D, DQ, 0);
    gemm(SLOTS, D, W_V, nullptr, VALS, B * R, D, DV, 0);

    // 8) query; attention; mem_read
    pack(seg2(HAVG, D, D, ctx, DC, DC), D + DC);
    gemm(XPAD, ceil4(D + DC), rq_w, rq_b, QRY, B, D + DC, DQ, 0);
    attention_k<<<B, 128, 0, stream>>>(QRY, KEYS, VALS, MRV);
    gemm(MRV, DV, vup_w, vup_b, MRD, B, DV, D, 0);

    // 9) down/up with modded weights -> output
    down_proj<<<B, DP, 0, stream>>>(MRD, wdb, wdm, DOWN);
    up_out<<<dim3(D / 256, B), 256, 0, stream>>>(DOWN, wub, wum, MRD, OUTV);

    // 10) pattern/address generators from pwi (pack once, two users)
    pack(seg4v(HAVG, D, D, SURP, 1, 1, MRD, D, D, ctx, DC, DC), PWIN);
    gemm(XPAD, ceil4(PWIN), pd1_w, pd1_b, TBUF, B, PWIN, DH, 1);
    gemm(TBUF, DH, pd2_w, pd2_b, RAWD, B, DH, D + DP, 0);
    gemm(XPAD, ceil4(PWIN), pu1_w, pu1_b, TBUF, B, PWIN, DH, 1);
    gemm(TBUF, DH, pu2_w, pu2_b, RAWU, B, DH, DP + D, 0);

    // 11) plr, sc, rank-1 fast-weight updates (write outputs directly)
    pack(seg2(SURP, 1, 1, ctx, DC, DC), 1 + DC);
    gemm(XPAD, ceil4(1 + DC), pl1_w, pl1_b, TBUF, B, 1 + DC, 64, 1);
    gemm(TBUF, 64, pl2_w, pl2_b, PLR, B, 64, 1, 3);
    calc_sc<<<1, 32, 0, stream>>>(SURP, PLR, SCb);
    wdn_k<<<(unsigned)(WDN_N / 256), 256, 0, stream>>>(wdm, SCb, RAWD, wdnout);
    wun_k<<<(unsigned)(WUN_N / 256), 256, 0, stream>>>(wum, SCb, RAWU, wunout);

    // 12) gate g = sigmoid(mlp([h_avg, output, context], g1, g2))
    pack(seg3(HAVG, D, D, OUTV, D, D, ctx, DC, DC), D + D + DC);
    gemm(XPAD, ceil4(D + D + DC), g1_w, g1_b, TBUF, B, D + D + DC, DH, 1);
    gemm(TBUF, DH, g2_w, g2_b, Gb, B, DH, 1, 2);

    // 13) h_new = LN(h + g*output) — overwrites the whole h_new region,
    //     including the (now dead) PARTS/SLOTS/KEYS/VALS scratch.
    final_ln<<<B * S, 256, 0, stream>>>(h, OUTV, Gb, ln_w, ln_b, hnew);
}